// ConcatPassageReranker_72610717106203
// MI455X (gfx1250) — compile-verified
//
#include <hip/hip_runtime.h>

// Problem constants (from the reference).
#define B_ 16
#define S_ 4096
#define H_ 1024
#define P_ 64
#define TITLE_ID_ 1
#define CTX_ID_ 2

#define NWAVE_ 8
#define KCHUNK_ (H_ / NWAVE_)   // 128 K-elements per wave per segment

typedef __attribute__((ext_vector_type(2))) float v2f;
typedef __attribute__((ext_vector_type(8))) float v8f;

// ---------------------------------------------------------------------------
// Phase 1: per-sample stable packed scan of token positions.
// One wave (32 lanes) per sample; ballot + prefix-popcount keeps positions in
// ascending order, matching the reference's sorted packing.
// ---------------------------------------------------------------------------
__global__ __launch_bounds__(32) void scan_tokens(const int* __restrict__ ids,
                                                  int* __restrict__ posT,
                                                  int* __restrict__ posC,
                                                  int* __restrict__ nT,
                                                  int* __restrict__ nC) {
  const int b = blockIdx.x;
  const int lane = threadIdx.x;
  const int* row = ids + (size_t)b * S_;
  const unsigned lt = (1u << lane) - 1u;
  int cntT = 0, cntC = 0;
  for (int base = 0; base < S_; base += 32) {
    const int id = row[base + lane];
    const unsigned mT = __builtin_amdgcn_ballot_w32(id == TITLE_ID_);
    const unsigned mC = __builtin_amdgcn_ballot_w32(id == CTX_ID_);
    if (id == TITLE_ID_) {
      const int idx = cntT + __builtin_popcount(mT & lt);
      if (idx < P_) posT[b * P_ + idx] = base + lane;
    }
    if (id == CTX_ID_) {
      const int idx = cntC + __builtin_popcount(mC & lt);
      if (idx < P_) posC[b * P_ + idx] = base + lane;
    }
    cntT += __builtin_popcount(mT);
    cntC += __builtin_popcount(mC);
  }
  if (lane == 0) {
    nT[b] = cntT < P_ ? cntT : P_;
    nC[b] = cntC < P_ ? cntC : P_;
  }
}

// ---------------------------------------------------------------------------
// Phase 2: scores via V_WMMA_F32_16X16X4_F32 (exact f32 math).
//
// Block = 256 threads (8 waves) handling one tile of 16 passages of one
// sample. All waves walk the same 3 segments (cls row / title row / ctx row)
// but each owns a fixed 128-wide K-chunk (kbase = wave*128) inside every
// segment, so every loop has a literal trip count of 32 and fully unrolls
// into  global_load_b64 (imm-offset)  clauses + back-to-back WMMAs.
// `wave` is forced into an SGPR via readfirstlane so all control flow and
// addressing is wave-uniform -> EXEC stays all-ones through every WMMA
// (architectural requirement).
//
// A layout (16x4 f32): lane l -> row m = l&15, K-pair 2*(l>>4); each lane
// loads row[k + 2*hi .. +1] as one b64. B is the weight vector broadcast to
// all 16 columns (identical per-lane addressing from vt_w), so every column
// of D holds the score vector; lanes 0/16 own rows 0-7 / 8-15 across the 8
// accumulator VGPRs. Partial D's from the 8 waves are summed through LDS.
// ---------------------------------------------------------------------------
__global__ __launch_bounds__(256) void wmma_score(const float* __restrict__ outputs,
                                                  const float* __restrict__ vt_w,
                                                  const int* __restrict__ posT,
                                                  const int* __restrict__ posC,
                                                  const int* __restrict__ nT,
                                                  const int* __restrict__ nC,
                                                  float* __restrict__ out) {
  __shared__ float red[NWAVE_][16];

  const int b    = blockIdx.x >> 2;   // 4 tiles of 16 passages per sample
  const int tile = blockIdx.x & 3;
  const int tid  = threadIdx.x;
  const int wave = __builtin_amdgcn_readfirstlane(tid >> 5);  // SGPR-uniform
  const int lane = tid & 31;
  const int mlo  = lane & 15;         // row within 16-passage tile
  const int hi   = lane >> 4;         // K-pair selector (0 -> K0..1, 1 -> K2..3)
  const int m    = tile * 16 + mlo;   // passage index

  const int nTv = nT[b];
  const int nCv = nC[b];
  const int validT = (m < nTv) ? 1 : 0;
  const int validC = (m < nCv) ? 1 : 0;
  // Clamp invalid rows to row 0 (workspace is poisoned). Title rows get an
  // explicit zero mask (reference zeroes ttl by its own validity); cls/ctx
  // invalidity is subsumed by the final -inf bias.
  const int rT = validT ? posT[b * P_ + m] : 0;
  const int rC = validC ? posC[b * P_ + m] : 0;

  const int kbase = wave * KCHUNK_;            // scalar: 0,128,...,896
  const size_t rowB = (size_t)b * S_;

  v8f c = {0.f, 0.f, 0.f, 0.f, 0.f, 0.f, 0.f, 0.f};

  // ---- segment 0: CLS row (row 0 of the sample), weights w[0:H) ----
  {
    const float* a = outputs + rowB * H_ + kbase + 2 * hi;
    const float* w = vt_w + kbase + 2 * hi;
    #pragma unroll
    for (int k = 0; k < KCHUNK_; k += 4) {
      v2f av = *(const v2f*)(a + k);
      v2f wv = *(const v2f*)(w + k);
      c = __builtin_amdgcn_wmma_f32_16x16x4_f32(false, av, false, wv,
                                                (short)0, c, false, false);
    }
  }

  // ---- segment 1: title row, weights w[H:2H), zero-masked per row ----
  {
    const float* a = outputs + (rowB + (size_t)rT) * H_ + kbase + 2 * hi;
    const float* w = vt_w + H_ + kbase + 2 * hi;
    const float msk = validT ? 1.0f : 0.0f;
    #pragma unroll
    for (int k = 0; k < KCHUNK_; k += 4) {
      v2f av = *(const v2f*)(a + k);
      v2f wv = *(const v2f*)(w + k);
      av *= msk;
      c = __builtin_amdgcn_wmma_f32_16x16x4_f32(false, av, false, wv,
                                                (short)0, c, false, false);
    }
  }

  // ---- segment 2: ctx row, weights w[2H:3H) (-inf covers invalid) ----
  {
    const float* a = outputs + (rowB + (size_t)rC) * H_ + kbase + 2 * hi;
    const float* w = vt_w + 2 * H_ + kbase + 2 * hi;
    #pragma unroll
    for (int k = 0; k < KCHUNK_; k += 4) {
      v2f av = *(const v2f*)(a + k);
      v2f wv = *(const v2f*)(w + k);
      c = __builtin_amdgcn_wmma_f32_16x16x4_f32(false, av, false, wv,
                                                (short)0, c, false, false);
    }
  }

  // D layout: lane 0 holds rows 0-7 in c[0..7], lane 16 holds rows 8-15.
  if (mlo == 0) {
    #pragma unroll
    for (int j = 0; j < 8; ++j) red[wave][hi * 8 + j] = c[j];
  }
  __syncthreads();

  if (tid < 16) {
    float s = 0.0f;
    #pragma unroll
    for (int w = 0; w < NWAVE_; ++w) s += red[w][tid];
    const int gm = tile * 16 + tid;
    const float bias = (gm < nCv) ? 0.0f : -__builtin_inff();
    out[b * P_ + gm] = s + bias;
  }
}

// ---------------------------------------------------------------------------
// Inputs (setup_inputs order): 0=outputs f32[B,S,H], 1=vt_w f32[3H],
// 2=input_ids i32[B,S], 3=max_passages, 4=title_id, 5=ctx_id (scalars,
// hardcoded per the reference constants). Output: f32[B,P].
// ---------------------------------------------------------------------------
extern "C" void kernel_launch(void* const* d_in, const int* in_sizes, int n_in,
                              void* d_out, int out_size, void* d_ws, size_t ws_size,
                              hipStream_t stream) {
  (void)in_sizes; (void)n_in; (void)out_size; (void)ws_size;
  const float* outputs = (const float*)d_in[0];
  const float* vt_w    = (const float*)d_in[1];
  const int*   ids     = (const int*)d_in[2];

  char* ws  = (char*)d_ws;
  int* posT = (int*)ws;                                   // 16*64 ints
  int* posC = (int*)(ws + B_ * P_ * sizeof(int));         // 16*64 ints
  int* nT   = (int*)(ws + 2 * B_ * P_ * sizeof(int));     // 16 ints
  int* nC   = nT + B_;                                    // 16 ints

  scan_tokens<<<B_, 32, 0, stream>>>(ids, posT, posC, nT, nC);
  wmma_score<<<B_ * (P_ / 16), 256, 0, stream>>>(outputs, vt_w, posT, posC, nT, nC,
                                                 (float*)d_out);
}